// DyGFormer_1889785610786
// MI455X (gfx1250) — compile-verified
//
#include <hip/hip_runtime.h>
#include <hip/hip_bf16.h>

// ---------------------------------------------------------------------------
// DyGFormer neighbor co-occurrence encoder, collapsed to:
//   G[c] = relu(c*W1 + b1) @ W2 + b2          (513 x 128 lookup table, WMMA)
//   out_src[b,j,:] = G[c0(src)] + G[c1(src)]  (histogram + table gather)
//   out_dst[b,j,:] = G[c0(dst)] + G[c1(dst)]
// Output writes (134 MB) dominate -> bandwidth-bound; table GEMM uses
// V_WMMA_F32_16X16X4_F32 (fp32, exact).
// ---------------------------------------------------------------------------

typedef __attribute__((ext_vector_type(2))) float v2f;
typedef __attribute__((ext_vector_type(8))) float v8f;

#define BATCH 256
#define LSEQ 512
#define DDIM 128
#define NNODES 1024
#define MAXC 512   // counts range 0..512 inclusive -> 513 table rows

// ---------------------------------------------------------------------------
// Kernel 1: build lookup table G[c, d], c in 0..512, d in 0..127 using
// fp32 WMMA 16x16x4.  Grid: (33 M-tiles, 8 N-tiles), 1 wave (32 thr) each.
//
// A (16x4 f32) ISA layout: lanes 0-15 hold M=0..15 with K={0,1} in the two
// VGPRs; lanes 16-31 hold the same M rows with K={2,3}.
// B (4x16 f32): lanes hold N=lane&15; VGPRs hold the K pair selected by
// lane>>4 (mirror of A).
// C/D (16x16 f32, 8 VGPRs): VGPR i = row (i + 8*(lane>>4)), col = lane&15.
// ---------------------------------------------------------------------------
__global__ void __launch_bounds__(32)
dyg_table_wmma(const float* __restrict__ W1, const float* __restrict__ b1,
               const float* __restrict__ W2, const float* __restrict__ b2,
               float* __restrict__ G) {
  const int mtile = blockIdx.x;      // 0..32  (rows of c)
  const int ntile = blockIdx.y;      // 0..7   (cols of d)
  const int lane  = threadIdx.x;     // 0..31
  const int mrow  = lane & 15;
  const int half  = lane >> 4;
  const int ncol  = lane & 15;

  const float c = (float)(mtile * 16 + mrow);

  v8f acc = {};
  #pragma unroll 4
  for (int k0 = 0; k0 < DDIM; k0 += 4) {
    const int ka = k0 + 2 * half;     // this lane's K pair base
    v2f a, b;
    // A = relu(c * W1 + b1), generated in-register
    a.x = fmaxf(c * W1[ka]     + b1[ka],     0.0f);
    a.y = fmaxf(c * W1[ka + 1] + b1[ka + 1], 0.0f);
    // B = W2[k, ntile*16 + n]
    b.x = W2[(ka)     * DDIM + ntile * 16 + ncol];
    b.y = W2[(ka + 1) * DDIM + ntile * 16 + ncol];
    acc = __builtin_amdgcn_wmma_f32_16x16x4_f32(
        /*neg_a=*/false, a, /*neg_b=*/false, b,
        /*c_mod=*/(short)0, acc, /*reuse_a=*/false, /*reuse_b=*/false);
  }

  const float bb = b2[ntile * 16 + ncol];
  #pragma unroll
  for (int i = 0; i < 8; ++i) {
    const int row = mtile * 16 + i + 8 * half;
    if (row <= MAXC) {
      G[row * DDIM + ntile * 16 + ncol] = acc[i] + bb;
    }
  }
}

// ---------------------------------------------------------------------------
// Kernel 2: per-batch histogram (LDS atomics) + count resolution + table
// gather with float4 (b128) coalesced stores.  1 block (256 thr) per batch.
// ---------------------------------------------------------------------------
__global__ void __launch_bounds__(256)
dyg_encode(const int* __restrict__ src_ids, const int* __restrict__ dst_ids,
           const float* __restrict__ G,
           float* __restrict__ out_src, float* __restrict__ out_dst) {
  __shared__ int hist_s[NNODES];
  __shared__ int hist_d[NNODES];
  __shared__ int ids_s[LSEQ];
  __shared__ int ids_d[LSEQ];
  __shared__ int cs0[LSEQ], cs1[LSEQ], cd0[LSEQ], cd1[LSEQ];

  const int b = blockIdx.x;
  const int t = threadIdx.x;

  #pragma unroll
  for (int i = t; i < NNODES; i += 256) { hist_s[i] = 0; hist_d[i] = 0; }
  __syncthreads();

  const int* __restrict__ srow = src_ids + b * LSEQ;
  const int* __restrict__ drow = dst_ids + b * LSEQ;
  #pragma unroll
  for (int i = t; i < LSEQ; i += 256) {
    const int s = srow[i];
    const int d = drow[i];
    ids_s[i] = s;
    ids_d[i] = d;
    atomicAdd(&hist_s[s], 1);
    atomicAdd(&hist_d[d], 1);
  }
  __syncthreads();

  // resolve counts (zero where id == 0, matching reference masking)
  #pragma unroll
  for (int i = t; i < LSEQ; i += 256) {
    const int s = ids_s[i];
    const int d = ids_d[i];
    cs0[i] = (s == 0) ? 0 : hist_s[s];
    cs1[i] = (s == 0) ? 0 : hist_d[s];
    cd0[i] = (d == 0) ? 0 : hist_s[d];
    cd1[i] = (d == 0) ? 0 : hist_d[d];
  }
  __syncthreads();

  // gather + store: 32 threads cover one position's 128 floats as float4.
  const int q   = t & 31;        // float4 column within the D=128 row
  const int jof = t >> 5;        // 0..7
  const float4* __restrict__ G4 = (const float4*)G;  // row stride = 32 float4
  float4* __restrict__ os = (float4*)(out_src + (size_t)b * LSEQ * DDIM);
  float4* __restrict__ od = (float4*)(out_dst + (size_t)b * LSEQ * DDIM);

  for (int j = jof; j < LSEQ; j += 8) {
    const float4 a0 = G4[cs0[j] * 32 + q];
    const float4 a1 = G4[cs1[j] * 32 + q];
    const float4 b0 = G4[cd0[j] * 32 + q];
    const float4 b1v = G4[cd1[j] * 32 + q];
    float4 rs, rd;
    rs.x = a0.x + a1.x; rs.y = a0.y + a1.y; rs.z = a0.z + a1.z; rs.w = a0.w + a1.w;
    rd.x = b0.x + b1v.x; rd.y = b0.y + b1v.y; rd.z = b0.z + b1v.z; rd.w = b0.w + b1v.w;
    os[j * 32 + q] = rs;
    od[j * 32 + q] = rd;
  }
}

// ---------------------------------------------------------------------------
extern "C" void kernel_launch(void* const* d_in, const int* in_sizes, int n_in,
                              void* d_out, int out_size, void* d_ws, size_t ws_size,
                              hipStream_t stream) {
  const int*   src_ids = (const int*)d_in[0];    // (B, L) int32
  const int*   dst_ids = (const int*)d_in[1];    // (B, L) int32
  const float* W1      = (const float*)d_in[2];  // (1, D)
  const float* b1      = (const float*)d_in[3];  // (D,)
  const float* W2      = (const float*)d_in[4];  // (D, D)
  const float* b2      = (const float*)d_in[5];  // (D,)

  float* out_src = (float*)d_out;                           // (B, L, D)
  float* out_dst = (float*)d_out + (size_t)BATCH * LSEQ * DDIM;

  float* G = (float*)d_ws;                       // 513 x 128 fp32 = 257 KB
  (void)ws_size; (void)in_sizes; (void)n_in; (void)out_size;

  // Kernel 1: lookup table via fp32 WMMA. 33 M-tiles x 8 N-tiles, 1 wave each.
  dyg_table_wmma<<<dim3(33, 8), 32, 0, stream>>>(W1, b1, W2, b2, G);

  // Kernel 2: histogram + gather + streaming stores. One block per batch.
  dyg_encode<<<BATCH, 256, 0, stream>>>(src_ids, dst_ids, G, out_src, out_dst);
}